// DynamicAdjacencyLayer_69501160784249
// MI455X (gfx1250) — compile-verified
//
#include <hip/hip_runtime.h>
#include <math.h>

typedef __attribute__((ext_vector_type(2))) float v2f;
typedef __attribute__((ext_vector_type(8))) float v8f;

#define NB   8      // batches
#define NN   2048   // rows/cols
#define ND   128    // feature dim
#define CEPS 1e-8f

// ---------------------------------------------------------------------------
// Kernel 1: reciprocal row norms.  One wave32 per row (128 elems -> 4/lane).
// ---------------------------------------------------------------------------
__global__ void rnorm_kernel(const float* __restrict__ h, float* __restrict__ rnorm) {
    const int row  = blockIdx.x * 8 + (threadIdx.x >> 5);   // 8 rows per 256-thr block
    const int lane = threadIdx.x & 31;
    const float* hr = h + (size_t)row * ND;
    float s = 0.f;
#pragma unroll
    for (int i = 0; i < 4; ++i) {
        float v = hr[lane + 32 * i];
        s += v * v;
    }
#pragma unroll
    for (int off = 16; off; off >>= 1) s += __shfl_xor(s, off, 32);
    if (lane == 0) rnorm[row] = 1.0f / fmaxf(sqrtf(s), CEPS);
}

// ---------------------------------------------------------------------------
// Kernel 2: fused  sim = (h·hT)*rn_i*rn_j  ->  softmax(sim/T)  -> out.
// One workgroup (8 wave32) per (batch, 16-row block).  The 16x2048 f32 sim
// slab lives in 128 KB of dynamic LDS (WGP has 320 KB) so the similarity
// matrix never round-trips through HBM.
// ---------------------------------------------------------------------------
__global__ void __launch_bounds__(256, 1)
simsoftmax_kernel(const float* __restrict__ h,
                  const float* __restrict__ rnorm,
                  const float* __restrict__ temp,
                  float* __restrict__ out) {
    extern __shared__ float lds[];                 // 16 * 2048 floats = 128 KB

    const int b       = blockIdx.x >> 7;           // 8 batches
    const int rb      = blockIdx.x & 127;          // 128 row-blocks of 16
    const int wave    = threadIdx.x >> 5;          // 0..7
    const int lane    = threadIdx.x & 31;
    const int n       = lane & 15;                 // M for A-frag, N for B-frag
    const int khalf   = lane >> 4;                 // which K pair within chunk
    const int rowbase = rb * 16;

    const float* hb = h     + (size_t)b * NN * ND;
    const float* rn = rnorm + b * NN;

    // ---- preload A fragments: this block's 16 rows, full K=128 ------------
    // f32 16x4 A layout: lane(m = lane&15); VGPR0 = K{0|2}, VGPR1 = K{1|3}
    // -> per lane a contiguous float2 at k0 + 2*khalf.
    v2f afrag[32];
    const float* arow = hb + (size_t)(rowbase + n) * ND + 2 * khalf;
#pragma unroll
    for (int kc = 0; kc < 32; ++kc)
        afrag[kc] = *(const v2f*)(arow + kc * 4);

    float rowr[8];                                 // rnorm of my 8 output rows
#pragma unroll
    for (int r = 0; r < 8; ++r)
        rowr[r] = rn[rowbase + r + 8 * khalf];

    // ---- GEMM over column tiles (8 waves stride the 128 tiles) ------------
    for (int t = wave; t < 128; t += 8) {
        const int colbase = t * 16;
        const float* brow = hb + (size_t)(colbase + n) * ND + 2 * khalf;
        v8f acc = {};
#pragma unroll
        for (int kc = 0; kc < 32; ++kc) {
            v2f bfrag = *(const v2f*)(brow + kc * 4);
            acc = __builtin_amdgcn_wmma_f32_16x16x4_f32(
                false, afrag[kc], false, bfrag, (short)0, acc, false, false);
        }
        // C/D layout: VGPR r, lanes0-15 -> M=r, lanes16-31 -> M=r+8; N=lane&15
        const float rc = rn[colbase + n];
#pragma unroll
        for (int r = 0; r < 8; ++r)
            lds[(r + 8 * khalf) * NN + colbase + n] = acc[r] * rowr[r] * rc;
    }

    __syncthreads();

    // ---- softmax: each wave owns 2 of the 16 rows -------------------------
    const float invT = 1.0f / temp[0];
#pragma unroll
    for (int rr = 0; rr < 2; ++rr) {
        const int row = wave * 2 + rr;
        float* srow = lds + row * NN;

        float m = -INFINITY;
        for (int c = lane; c < NN; c += 32) m = fmaxf(m, srow[c]);
#pragma unroll
        for (int off = 16; off; off >>= 1) m = fmaxf(m, __shfl_xor(m, off, 32));

        float s = 0.f;
        for (int c = lane; c < NN; c += 32) {
            float e = __expf((srow[c] - m) * invT);
            srow[c] = e;
            s += e;
        }
#pragma unroll
        for (int off = 16; off; off >>= 1) s += __shfl_xor(s, off, 32);

        const float inv = 1.0f / s;
        float* orow = out + ((size_t)b * NN + rowbase + row) * NN;
        for (int c = lane; c < NN; c += 32) orow[c] = srow[c] * inv;  // coalesced
    }
}

// ---------------------------------------------------------------------------
extern "C" void kernel_launch(void* const* d_in, const int* in_sizes, int n_in,
                              void* d_out, int out_size, void* d_ws, size_t ws_size,
                              hipStream_t stream) {
    const float* h    = (const float*)d_in[0];   // [8,2048,128] f32
    const float* temp = (const float*)d_in[1];   // scalar
    float*       out  = (float*)d_out;           // [8,2048,2048] f32
    float*       rnrm = (float*)d_ws;            // 16384 floats (64 KB)

    rnorm_kernel<<<NB * NN / 8, 256, 0, stream>>>(h, rnrm);
    simsoftmax_kernel<<<NB * (NN / 16), 256, 16 * NN * sizeof(float), stream>>>(
        h, rnrm, temp, out);
}